// Sampler_36043365548486
// MI455X (gfx1250) — compile-verified
//
#include <hip/hip_runtime.h>
#include <hip/hip_bf16.h>
#include <stdint.h>

#define BATCH   256
#define VOCAB   128000
#define THREADS 512
#define NWAVES  (THREADS / 32)
#define TILE    2000           // floats per tile (8000 B); threads 0..499 carry 16 B each
#define DEPTH   4              // async pipeline depth (ring of 4 tiles in flight)
#define NTILES  (VOCAB / TILE) // 64, exact -> uniform ASYNCcnt across all waves
#define NBINS   2048
#define RANGE   32.0f          // exp(-32) ~ 1e-14 : negligible mass below m-RANGE

// ---------------- CDNA5 async global->LDS helpers (ASYNCcnt tracked) ----------------

__device__ __forceinline__ void issue_tile(const float* rowp, int tileIdx, float* buf, int tid) {
  // Every tile: threads 0..499 active => every wave (incl. wave 15, partial EXEC)
  // issues exactly one async op per tile, so ASYNCcnt is uniform across waves.
  if (tid * 4 < TILE) {
    uint32_t lds = (uint32_t)(uintptr_t)buf + (uint32_t)(tid * 16);   // LDS byte address
    uint32_t off = (uint32_t)(tileIdx * TILE + tid * 4) * 4u;         // byte offset in row
    // GVS mode: reads global memory at the 64-bit base rowp plus the 32-bit byte offset off; writes LDS at lds
    asm volatile("global_load_async_to_lds_b128 %0, %1, %2"
                 :: "v"(lds), "v"(off), "s"(rowp)
                 : "memory");
  }
}

// Stream one row through a DEPTH-deep LDS ring. Each thread consumes only the
// 16 B chunk it fetched itself, so no workgroup barriers are needed inside.
template <typename F>
__device__ __forceinline__ void stream_row(const float* rowp, float* ring, int tid, F&& f) {
  #pragma unroll
  for (int k = 0; k < DEPTH; ++k)
    issue_tile(rowp, k, ring + k * TILE, tid);
  for (int k = 0; k < NTILES; ++k) {
    const int rem = NTILES - 1 - k;       // tiles still in flight beyond tile k
    if (rem >= 3)      asm volatile("s_wait_asynccnt 0x3" ::: "memory");
    else if (rem == 2) asm volatile("s_wait_asynccnt 0x2" ::: "memory");
    else if (rem == 1) asm volatile("s_wait_asynccnt 0x1" ::: "memory");
    else               asm volatile("s_wait_asynccnt 0x0" ::: "memory");
    float* cur = ring + (k & (DEPTH - 1)) * TILE;
    f(cur, k * TILE);
    const int nk = k + DEPTH;
    if (nk < NTILES) issue_tile(rowp, nk, cur, tid);
  }
}

// ---------------- reductions (wave32) ----------------

// argmax of (v, tie -> smaller i), carrying payload x; broadcasts result to all threads
__device__ __forceinline__ void block_argmax3(float& v, int& i, float& x,
                                              float* rf, int* ri, float* rx, int tid) {
  #pragma unroll
  for (int o = 16; o > 0; o >>= 1) {
    float ov = __shfl_down(v, o);
    int   oi = __shfl_down(i, o);
    float ox = __shfl_down(x, o);
    if (ov > v || (ov == v && oi < i)) { v = ov; i = oi; x = ox; }
  }
  if ((tid & 31) == 0) { rf[tid >> 5] = v; ri[tid >> 5] = i; rx[tid >> 5] = x; }
  __syncthreads();
  if (tid == 0) {
    for (int w = 1; w < NWAVES; ++w)
      if (rf[w] > v || (rf[w] == v && ri[w] < i)) { v = rf[w]; i = ri[w]; x = rx[w]; }
    rf[0] = v; ri[0] = i; rx[0] = x;
  }
  __syncthreads();
  v = rf[0]; i = ri[0]; x = rx[0];
  __syncthreads();   // safe to reuse scratch afterwards
}

__device__ __forceinline__ uint32_t hashmix(uint32_t a) {
  a ^= a >> 16; a *= 0x7feb352dU;
  a ^= a >> 15; a *= 0x846ca68bU;
  a ^= a >> 16;
  return a;
}

// ---------------- kernel: one block per batch row ----------------

__global__ void __launch_bounds__(THREADS)
topp_sampler_kernel(const float* __restrict__ logits,
                    const float* __restrict__ temps,
                    const float* __restrict__ top_ps,
                    float* __restrict__ out) {
  __shared__ __align__(16) float ring[DEPTH * TILE];   // 31.25 KB async ring
  __shared__ float hist[NBINS];                        // 8 KB
  __shared__ float rf[NWAVES];
  __shared__ int   ri[NWAVES];
  __shared__ float rx[NWAVES];
  __shared__ float s_tau, s_keepZ;

  const int row = blockIdx.x;
  const int tid = threadIdx.x;
  const float* rowp = logits + (size_t)row * VOCAB;

  const float t     = temps[row];
  const float inv_t = (t == 0.0f) ? 1.0f : (1.0f / t);   // safe_t semantics

  // ---------- Pass A: max + argmax of scaled logits (HBM -> L2) ----------
  float m = -INFINITY; int greedy = 0x7fffffff; float dummy = 0.0f;
  stream_row(rowp, ring, tid, [&](float* cur, int start) {
    if (tid * 4 < TILE) {
      float4 v = *reinterpret_cast<float4*>(&cur[tid * 4]);
      float xs[4] = {v.x, v.y, v.z, v.w};
      #pragma unroll
      for (int j = 0; j < 4; ++j) {
        float x = xs[j] * inv_t;
        int idx = start + tid * 4 + j;
        if (x > m || (x == m && idx < greedy)) { m = x; greedy = idx; }
      }
    }
  });
  block_argmax3(m, greedy, dummy, rf, ri, rx, tid);   // all threads now hold (m, greedy)

  // ---------- Pass B: exp-mass histogram over [m-RANGE, m] (L2 resident) ----------
  for (int i = tid; i < NBINS; i += THREADS) hist[i] = 0.0f;
  __syncthreads();

  const float lo    = m - RANGE;
  const float invbw = (float)NBINS / RANGE;
  stream_row(rowp, ring, tid, [&](float* cur, int start) {
    if (tid * 4 < TILE) {
      float4 v = *reinterpret_cast<float4*>(&cur[tid * 4]);
      float xs[4] = {v.x, v.y, v.z, v.w};
      #pragma unroll
      for (int j = 0; j < 4; ++j) {
        float x = xs[j] * inv_t;
        if (x >= lo) {                       // below lo: < 1e-14 relative mass, drop
          int b = (int)((x - lo) * invbw);
          if (b > NBINS - 1) b = NBINS - 1;
          atomicAdd(&hist[b], __expf(x - m));
        }
      }
    }
  });
  __syncthreads();

  // total mass Z (== softmax denominator up to factor exp(m))
  float zpart = 0.0f;
  for (int i = tid; i < NBINS; i += THREADS) zpart += hist[i];
  #pragma unroll
  for (int o = 16; o > 0; o >>= 1) zpart += __shfl_down(zpart, o);
  if ((tid & 31) == 0) rf[tid >> 5] = zpart;
  __syncthreads();

  if (tid == 0) {
    float Z = 0.0f;
    for (int w = 0; w < NWAVES; ++w) Z += rf[w];
    float p = top_ps[row];
    p = fminf(fmaxf(p, 0.0f), 1.0f);
    const float target = p * Z;
    // scan bins top-down; cut inside boundary bin (boundary bin kept -> top-1 kept)
    float acc = 0.0f;
    int b = NBINS;
    while (b > 0) {
      --b;
      acc += hist[b];
      if (acc > target) break;              // early exit: mass concentrates near m
    }
    s_tau   = lo + (float)b * (RANGE / (float)NBINS);
    s_keepZ = fmaxf(acc, 1e-30f);
  }
  __syncthreads();

  const float tau   = s_tau;
  const float keepZ = s_keepZ;

  // ---------- Pass C: Gumbel-max categorical over kept set (L2 resident) ----------
  float bs = -INFINITY; int bi = 0x7fffffff; float bx = 0.0f;
  stream_row(rowp, ring, tid, [&](float* cur, int start) {
    if (tid * 4 < TILE) {
      float4 v = *reinterpret_cast<float4*>(&cur[tid * 4]);
      float xs[4] = {v.x, v.y, v.z, v.w};
      #pragma unroll
      for (int j = 0; j < 4; ++j) {
        float x = xs[j] * inv_t;
        if (x >= tau) {
          int idx = start + tid * 4 + j;
          uint32_t h = hashmix((uint32_t)idx * 2654435761u
                               + (uint32_t)row * 40503u + 0x9E3779B9u);
          float u = ((float)h + 0.5f) * (1.0f / 4294967296.0f);   // (0,1)
          float g = -__logf(-__logf(u));                          // Gumbel(0,1)
          float s = x + g;
          if (s > bs || (s == bs && idx < bi)) { bs = s; bi = idx; bx = x; }
        }
      }
    }
  });
  block_argmax3(bs, bi, bx, rf, ri, rx, tid);

  if (tid == 0) {
    const int   token = (t == 0.0f) ? greedy : bi;
    const float sel   = (t == 0.0f) ? m      : bx;   // filt[token] (greedy token has x == m)
    const float logz  = m + __logf(keepZ);           // logsumexp of filtered logits
    out[row]         = (float)token;                 // tokens, flat first
    out[BATCH + row] = sel - logz;                   // log_probs second
  }
}

// ---------------- harness entry ----------------

extern "C" void kernel_launch(void* const* d_in, const int* in_sizes, int n_in,
                              void* d_out, int out_size, void* d_ws, size_t ws_size,
                              hipStream_t stream) {
  (void)in_sizes; (void)n_in; (void)out_size; (void)d_ws; (void)ws_size;
  const float* logits = (const float*)d_in[0];
  const float* temps  = (const float*)d_in[1];
  const float* tps    = (const float*)d_in[2];
  float* out = (float*)d_out;
  topp_sampler_kernel<<<BATCH, THREADS, 0, stream>>>(logits, temps, tps, out);
}